// GATFraudDetector_19567871001356
// MI455X (gfx1250) — compile-verified
//
#include <hip/hip_runtime.h>

// ---------------------------------------------------------------------------
// GAT fraud detector forward (3 GATConv layers + BN + ELU), gfx1250 / wave32.
// N=100000 nodes, E=1.6M edges (+N self loops), IN=HC=128, H=4 heads, C=32.
// ---------------------------------------------------------------------------

typedef float v2f __attribute__((ext_vector_type(2)));
typedef float v8f __attribute__((ext_vector_type(8)));

#define HC 128
#define NHEAD 4
#define CH 32

// ---- monotone float atomic max (works with -inf init) ----------------------
__device__ __forceinline__ void atomicMaxF(float* addr, float val) {
  if (val >= 0.0f) {
    atomicMax((int*)addr, __float_as_int(val));
  } else {
    atomicMin((unsigned int*)addr, (unsigned int)__float_as_int(val));
  }
}

// ---- generic fill ----------------------------------------------------------
__global__ void gat_fill_f32(float* __restrict__ p, float v, int n) {
  int i = blockIdx.x * blockDim.x + threadIdx.x;
  if (i < n) p[i] = v;
}

// ---- H = X @ W   (M x 128) @ (128 x 128), fp32 WMMA -----------------------
// Block: 256 threads = 8 waves. Block b covers rows [16b, 16b+16).
// Wave w computes the 16x16 tile at columns [16w, 16w+16).
// A frag (16x4 f32, 2 VGPR): lanes 0-15 -> M=lane, K={k0,k0+1};
//                            lanes 16-31 -> M=lane-16, K={k0+2,k0+3}.
// B frag (4x16 f32, 2 VGPR): lanes 0-15 -> N=lane, K={k0,k0+1};
//                            lanes 16-31 -> N=lane-16, K={k0+2,k0+3}.
// C/D (16x16 f32, 8 VGPR): elem(M = r + 8*(lane>>4), N = lane&15) in VGPR r.
__global__ __launch_bounds__(256) void gat_gemm128_wmma(
    const float* __restrict__ X, const float* __restrict__ W,
    float* __restrict__ Hout) {
  const int wave  = threadIdx.x >> 5;          // 0..7 -> column tile
  const int lane  = threadIdx.x & 31;
  const int l15   = lane & 15;
  const int khalf = (lane >> 4) << 1;          // 0 or 2
  const int row   = (blockIdx.x << 4) + l15;   // A-matrix row for this lane
  const int col   = (wave << 4) + l15;         // B/C/D column for this lane

  const float* __restrict__ Arow = X + (size_t)row * HC;

  v8f acc = {};
#pragma unroll
  for (int k0 = 0; k0 < HC; k0 += 4) {
    const int kk = k0 + khalf;
    v2f a;
    a.x = Arow[kk];
    a.y = Arow[kk + 1];
    v2f b;
    b.x = W[(size_t)kk * HC + col];
    b.y = W[(size_t)(kk + 1) * HC + col];
    acc = __builtin_amdgcn_wmma_f32_16x16x4_f32(
        /*neg_a=*/false, a, /*neg_b=*/false, b,
        /*c_mod=*/(short)0, acc, /*reuse_a=*/false, /*reuse_b=*/false);
  }

  const int rbase = (blockIdx.x << 4) + ((lane >> 4) << 3);
#pragma unroll
  for (int r = 0; r < 8; ++r) {
    Hout[(size_t)(rbase + r) * HC + col] = acc[r];
  }
}

// ---- per-node attention logits: als[n,h] = <h[n,h,:], a_s[h,:]> ------------
__global__ void gat_node_alpha(const float* __restrict__ H,
                               const float* __restrict__ a_s,
                               const float* __restrict__ a_d,
                               float* __restrict__ als,
                               float* __restrict__ ald, int n) {
  int node = blockIdx.x * blockDim.x + threadIdx.x;
  if (node >= n) return;
  const float4* hp = (const float4*)(H + (size_t)node * HC);
  const float4* as4 = (const float4*)a_s;
  const float4* ad4 = (const float4*)a_d;
#pragma unroll
  for (int h = 0; h < NHEAD; ++h) {
    float s = 0.f, d = 0.f;
#pragma unroll
    for (int j = 0; j < CH / 4; ++j) {
      float4 hv = hp[h * 8 + j];
      float4 av = as4[h * 8 + j];
      float4 bv = ad4[h * 8 + j];
      s += hv.x * av.x + hv.y * av.y + hv.z * av.z + hv.w * av.w;
      d += hv.x * bv.x + hv.y * bv.y + hv.z * bv.z + hv.w * bv.w;
    }
    als[node * NHEAD + h] = s;
    ald[node * NHEAD + h] = d;
  }
}

// ---- edge pass 1: segment max of leaky_relu logits (4 heads) ---------------
__global__ void gat_edge_max(const long long* __restrict__ ei, int E, int Nn,
                             const float* __restrict__ als,
                             const float* __restrict__ ald,
                             float* __restrict__ emax) {
  int e = blockIdx.x * blockDim.x + threadIdx.x;
  if (e >= E + Nn) return;
  int s, d;
  if (e < E) { s = (int)ei[e]; d = (int)ei[(size_t)E + e]; }
  else       { s = e - E; d = s; }
  const float4 sv = *(const float4*)(als + (size_t)s * NHEAD);
  const float4 dv = *(const float4*)(ald + (size_t)d * NHEAD);
  float v[NHEAD] = {sv.x + dv.x, sv.y + dv.y, sv.z + dv.z, sv.w + dv.w};
#pragma unroll
  for (int h = 0; h < NHEAD; ++h) {
    float t = v[h] > 0.f ? v[h] : 0.2f * v[h];
    atomicMaxF(&emax[(size_t)d * NHEAD + h], t);
  }
}

// ---- edge pass 2 (fused): den[d,h] += ee; agg[d,:] += ee * h[s,:] ----------
// One wave per edge. lane -> channels [4*lane, 4*lane+4), head = lane>>3.
__global__ __launch_bounds__(256) void gat_edge_agg(
    const long long* __restrict__ ei, int E, int Nn,
    const float* __restrict__ als, const float* __restrict__ ald,
    const float* __restrict__ emax, const float* __restrict__ H,
    float* __restrict__ den, float* __restrict__ agg) {
  int gid  = blockIdx.x * blockDim.x + threadIdx.x;
  int e    = gid >> 5;
  int lane = gid & 31;
  if (e >= E + Nn) return;
  int s, d;
  if (e < E) { s = (int)ei[e]; d = (int)ei[(size_t)E + e]; }
  else       { s = e - E; d = s; }
  const int head = lane >> 3;
  float v = als[(size_t)s * NHEAD + head] + ald[(size_t)d * NHEAD + head];
  v = v > 0.f ? v : 0.2f * v;
  const float ee = __expf(v - emax[(size_t)d * NHEAD + head]);
  if ((lane & 7) == 0) atomicAdd(&den[(size_t)d * NHEAD + head], ee);
  const float4 hv = ((const float4*)(H + (size_t)s * HC))[lane];
  float* ap = agg + (size_t)d * HC + lane * 4;
  atomicAdd(ap + 0, ee * hv.x);
  atomicAdd(ap + 1, ee * hv.y);
  atomicAdd(ap + 2, ee * hv.z);
  atomicAdd(ap + 3, ee * hv.w);
}

// ---- epilogue: normalize + bias + BN(eval) + ELU ---------------------------
__global__ void gat_epilogue_bn_elu(const float* __restrict__ agg,
                                    const float* __restrict__ den,
                                    const float* __restrict__ b,
                                    const float* __restrict__ g,
                                    const float* __restrict__ be,
                                    const float* __restrict__ m,
                                    const float* __restrict__ v,
                                    float* __restrict__ act, int n) {
  int i = blockIdx.x * blockDim.x + threadIdx.x;
  if (i >= n * HC) return;
  const int node = i >> 7;
  const int c    = i & (HC - 1);
  const int head = c >> 5;
  float x = agg[i] / (den[(size_t)node * NHEAD + head] + 1e-16f) + b[c];
  x = (x - m[c]) * rsqrtf(v[c] + 1e-5f) * g[c] + be[c];
  act[i] = x > 0.f ? x : (__expf(x) - 1.f);
}

// ---- layer 2: (N x 128) @ (128 x 2), fused attention logits ----------------
__global__ void gat_gemm_out2(const float* __restrict__ X,
                              const float* __restrict__ W2,
                              const float* __restrict__ as2,
                              const float* __restrict__ ad2,
                              float* __restrict__ h2,
                              float* __restrict__ als2,
                              float* __restrict__ ald2, int n) {
  int node = blockIdx.x * blockDim.x + threadIdx.x;
  if (node >= n) return;
  const float4* xp = (const float4*)(X + (size_t)node * HC);
  float o0 = 0.f, o1 = 0.f;
#pragma unroll 8
  for (int j = 0; j < HC / 4; ++j) {
    float4 x4 = xp[j];
    const float* w = W2 + (size_t)(4 * j) * 2;
    o0 += x4.x * w[0] + x4.y * w[2] + x4.z * w[4] + x4.w * w[6];
    o1 += x4.x * w[1] + x4.y * w[3] + x4.z * w[5] + x4.w * w[7];
  }
  h2[(size_t)node * 2 + 0] = o0;
  h2[(size_t)node * 2 + 1] = o1;
  als2[node] = o0 * as2[0] + o1 * as2[1];
  ald2[node] = o0 * ad2[0] + o1 * ad2[1];
}

__global__ void gat_edge_max1h(const long long* __restrict__ ei, int E, int Nn,
                               const float* __restrict__ als,
                               const float* __restrict__ ald,
                               float* __restrict__ emax) {
  int e = blockIdx.x * blockDim.x + threadIdx.x;
  if (e >= E + Nn) return;
  int s, d;
  if (e < E) { s = (int)ei[e]; d = (int)ei[(size_t)E + e]; }
  else       { s = e - E; d = s; }
  float v = als[s] + ald[d];
  v = v > 0.f ? v : 0.2f * v;
  atomicMaxF(&emax[d], v);
}

__global__ void gat_edge_agg1h(const long long* __restrict__ ei, int E, int Nn,
                               const float* __restrict__ als,
                               const float* __restrict__ ald,
                               const float* __restrict__ emax,
                               const float* __restrict__ h2,
                               float* __restrict__ den,
                               float* __restrict__ agg) {
  int e = blockIdx.x * blockDim.x + threadIdx.x;
  if (e >= E + Nn) return;
  int s, d;
  if (e < E) { s = (int)ei[e]; d = (int)ei[(size_t)E + e]; }
  else       { s = e - E; d = s; }
  float v = als[s] + ald[d];
  v = v > 0.f ? v : 0.2f * v;
  const float ee = __expf(v - emax[d]);
  atomicAdd(&den[d], ee);
  atomicAdd(&agg[(size_t)d * 2 + 0], ee * h2[(size_t)s * 2 + 0]);
  atomicAdd(&agg[(size_t)d * 2 + 1], ee * h2[(size_t)s * 2 + 1]);
}

__global__ void gat_final_out(const float* __restrict__ agg,
                              const float* __restrict__ den,
                              const float* __restrict__ b2,
                              float* __restrict__ out, int n) {
  int i = blockIdx.x * blockDim.x + threadIdx.x;
  if (i >= n * 2) return;
  out[i] = agg[i] / (den[i >> 1] + 1e-16f) + b2[i & 1];
}

// ---------------------------------------------------------------------------
static inline unsigned cdivu(long long a, long long b) {
  return (unsigned)((a + b - 1) / b);
}

extern "C" void kernel_launch(void* const* d_in, const int* in_sizes, int n_in,
                              void* d_out, int out_size, void* d_ws,
                              size_t ws_size, hipStream_t stream) {
  (void)n_in; (void)out_size; (void)ws_size;
  const float*     x   = (const float*)d_in[0];
  const long long* ei  = (const long long*)d_in[1];
  const float* W0  = (const float*)d_in[2];
  const float* as0 = (const float*)d_in[3];
  const float* ad0 = (const float*)d_in[4];
  const float* b0  = (const float*)d_in[5];
  const float* g0  = (const float*)d_in[6];
  const float* be0 = (const float*)d_in[7];
  const float* m0  = (const float*)d_in[8];
  const float* v0  = (const float*)d_in[9];
  const float* W1  = (const float*)d_in[10];
  const float* as1 = (const float*)d_in[11];
  const float* ad1 = (const float*)d_in[12];
  const float* b1  = (const float*)d_in[13];
  const float* g1  = (const float*)d_in[14];
  const float* be1 = (const float*)d_in[15];
  const float* m1  = (const float*)d_in[16];
  const float* v1  = (const float*)d_in[17];
  const float* W2  = (const float*)d_in[18];
  const float* as2 = (const float*)d_in[19];
  const float* ad2 = (const float*)d_in[20];
  const float* b2  = (const float*)d_in[21];

  const int Nn = in_sizes[0] / HC;     // 100000
  const int E  = in_sizes[1] / 2;      // 1600000
  const int E2 = E + Nn;               // edges incl. self loops

  // ---- workspace layout (all 128B aligned chunk sizes) ----
  char* ws = (char*)d_ws;
  float* Hbuf = (float*)ws;  ws += (size_t)Nn * HC * sizeof(float);
  float* AGG  = (float*)ws;  ws += (size_t)Nn * HC * sizeof(float);
  float* ACT  = (float*)ws;  ws += (size_t)Nn * HC * sizeof(float);
  float* ALS  = (float*)ws;  ws += (size_t)Nn * NHEAD * sizeof(float);
  float* ALD  = (float*)ws;  ws += (size_t)Nn * NHEAD * sizeof(float);
  float* EMX  = (float*)ws;  ws += (size_t)Nn * NHEAD * sizeof(float);
  float* DEN  = (float*)ws;  ws += (size_t)Nn * NHEAD * sizeof(float);
  float* H2   = (float*)ws;  ws += (size_t)Nn * 2 * sizeof(float);
  float* AGG2 = (float*)ws;  ws += (size_t)Nn * 2 * sizeof(float);
  float* ALS2 = (float*)ws;  ws += (size_t)Nn * sizeof(float);
  float* ALD2 = (float*)ws;  ws += (size_t)Nn * sizeof(float);
  float* EMX2 = (float*)ws;  ws += (size_t)Nn * sizeof(float);
  float* DEN2 = (float*)ws;  ws += (size_t)Nn * sizeof(float);

  const dim3 blk(256);
  const float NEG_BIG = -3.0e38f;

  const float* Wl[2]  = {W0, W1};
  const float* asl[2] = {as0, as1};
  const float* adl[2] = {ad0, ad1};
  const float* bl[2]  = {b0, b1};
  const float* gl[2]  = {g0, g1};
  const float* bel[2] = {be0, be1};
  const float* ml[2]  = {m0, m1};
  const float* vl[2]  = {v0, v1};

  for (int layer = 0; layer < 2; ++layer) {
    const float* Xin = (layer == 0) ? x : ACT;
    // H = Xin @ W  (WMMA fp32)
    gat_gemm128_wmma<<<Nn / 16, blk, 0, stream>>>(Xin, Wl[layer], Hbuf);
    // per-node logits
    gat_node_alpha<<<cdivu(Nn, 256), blk, 0, stream>>>(
        Hbuf, asl[layer], adl[layer], ALS, ALD, Nn);
    // init segment buffers
    gat_fill_f32<<<cdivu((long long)Nn * NHEAD, 256), blk, 0, stream>>>(
        EMX, NEG_BIG, Nn * NHEAD);
    gat_fill_f32<<<cdivu((long long)Nn * NHEAD, 256), blk, 0, stream>>>(
        DEN, 0.0f, Nn * NHEAD);
    gat_fill_f32<<<cdivu((long long)Nn * HC, 256), blk, 0, stream>>>(
        AGG, 0.0f, Nn * HC);
    // segment softmax + aggregation
    gat_edge_max<<<cdivu(E2, 256), blk, 0, stream>>>(ei, E, Nn, ALS, ALD, EMX);
    gat_edge_agg<<<cdivu((long long)E2 * 32, 256), blk, 0, stream>>>(
        ei, E, Nn, ALS, ALD, EMX, Hbuf, DEN, AGG);
    // normalize + bias + BN + ELU -> ACT
    gat_epilogue_bn_elu<<<cdivu((long long)Nn * HC, 256), blk, 0, stream>>>(
        AGG, DEN, bl[layer], gl[layer], bel[layer], ml[layer], vl[layer],
        ACT, Nn);
  }

  // ---- layer 2 (heads=1, out=2) ----
  gat_gemm_out2<<<cdivu(Nn, 256), blk, 0, stream>>>(ACT, W2, as2, ad2, H2,
                                                    ALS2, ALD2, Nn);
  gat_fill_f32<<<cdivu(Nn, 256), blk, 0, stream>>>(EMX2, NEG_BIG, Nn);
  gat_fill_f32<<<cdivu(Nn, 256), blk, 0, stream>>>(DEN2, 0.0f, Nn);
  gat_fill_f32<<<cdivu((long long)Nn * 2, 256), blk, 0, stream>>>(AGG2, 0.0f,
                                                                  Nn * 2);
  gat_edge_max1h<<<cdivu(E2, 256), blk, 0, stream>>>(ei, E, Nn, ALS2, ALD2,
                                                     EMX2);
  gat_edge_agg1h<<<cdivu(E2, 256), blk, 0, stream>>>(ei, E, Nn, ALS2, ALD2,
                                                     EMX2, H2, DEN2, AGG2);
  gat_final_out<<<cdivu((long long)Nn * 2, 256), blk, 0, stream>>>(
      AGG2, DEN2, b2, (float*)d_out, Nn);
}